// BlockAttentionResidual_22591527977642
// MI455X (gfx1250) — compile-verified
//
#include <hip/hip_runtime.h>
#include <hip/hip_bf16.h>

// Problem constants from the reference: blocks [L,B,T,D] fp32.
#define LQ 8
#define BQ 4
#define TQ 4096
#define DQ 1024
#define ROWP 1032  // padded LDS row pitch (floats): 1032 % 64 == 8 -> distinct banks per l-row

typedef __attribute__((ext_vector_type(2))) float v2f;
typedef __attribute__((ext_vector_type(8))) float v8f;

__launch_bounds__(256)
__global__ void BlockAttentionResidual_22591527977642_kernel(
    const float* __restrict__ blocks,   // [L,B,T,D]
    const float* __restrict__ query,    // [D]
    const float* __restrict__ normw,    // [D]
    float* __restrict__ out)            // [B,T,D]
{
    __shared__ __align__(16) float tile[LQ][ROWP];  // V[l, b, t, :] rows
    __shared__ __align__(16) float qw[DQ];          // query * norm_weight
    __shared__ float pdot[8][8];                    // [wave][l] partial dots
    __shared__ float sumsq[8];
    __shared__ float attn[8];

    const int tid  = threadIdx.x;
    const int lane = tid & 31;
    const int wave = tid >> 5;
    const int gid  = blockIdx.x;       // = b*T + t
    const int b    = gid >> 12;        // / 4096
    const int t    = gid & 4095;

    // ---------------- Phase 0: async fill of the 8 x 1024 fp32 tile ----------------
    // One global_load_async_to_lds_b128 per lane per l-row (256 lanes * 16B = 4KB row).
    const size_t base    = ((size_t)b * TQ + t) * (size_t)DQ + (size_t)tid * 4;
    const size_t lstride = (size_t)BQ * TQ * DQ;
#pragma unroll
    for (int l = 0; l < LQ; ++l) {
        unsigned long long ga =
            (unsigned long long)(uintptr_t)(blocks + base + (size_t)l * lstride);
        unsigned int loff =
            (unsigned int)(uintptr_t)(&tile[l][tid * 4]);  // low 32 bits = LDS byte offset
        asm volatile("global_load_async_to_lds_b128 %0, %1, off"
                     :: "v"(loff), "v"(ga) : "memory");
    }

    // qw = query * norm_weight (tiny, L2-resident) — regular path into LDS.
    {
        const float4 q = ((const float4*)query)[tid];
        const float4 w = ((const float4*)normw)[tid];
        *(float4*)&qw[tid * 4] = make_float4(q.x * w.x, q.y * w.y, q.z * w.z, q.w * w.w);
    }

    asm volatile("s_wait_asynccnt 0" ::: "memory");
    __syncthreads();

    // ---------------- Phase 1: sum of squares per l (wave w <-> l = w) ----------------
    {
        float s = 0.f;
#pragma unroll
        for (int j = 0; j < 8; ++j) {
            const float4 v = *(const float4*)&tile[wave][lane * 4 + j * 128];
            s += v.x * v.x + v.y * v.y + v.z * v.z + v.w * v.w;
        }
#pragma unroll
        for (int off = 16; off > 0; off >>= 1) s += __shfl_xor(s, off, 32);
        if (lane == 0) sumsq[wave] = s;
    }

    // ---------------- Phase 2: dot_l = sum_d qw[d]*V[l,d] via V_WMMA_F32_16X16X4_F32 --
    // A (16x4): row i<8 = V[l=i, d0+k], rows 8..15 zero. B (4x16): every column = qw[d0+k].
    // Each wave covers d in [wave*128, wave*128+128): 32 accumulating WMMAs.
#if defined(__HIP_DEVICE_COMPILE__)
    {
        const int  row = lane & 15;
        const int  rl  = row & 7;
        const bool act = row < 8;
        const int  kb  = (lane < 16) ? 0 : 2;   // K index pair held by this half-wave
        const int  dbase = wave * 128;
        v8f acc = {0.f, 0.f, 0.f, 0.f, 0.f, 0.f, 0.f, 0.f};
#pragma unroll
        for (int c = 0; c < 32; ++c) {
            const int d0 = dbase + c * 4;
            const float2 ta = *(const float2*)&tile[rl][d0 + kb];
            const float2 tb = *(const float2*)&qw[d0 + kb];
            v2f a;  a.x  = act ? ta.x : 0.f;  a.y  = act ? ta.y : 0.f;
            v2f bb; bb.x = tb.x;              bb.y = tb.y;
            acc = __builtin_amdgcn_wmma_f32_16x16x4_f32(
                false, a, false, bb, (short)0, acc, false, false);
        }
        // D layout: VGPR r, lane 0 = D[M=r][N=0]; columns identical by construction.
        if (lane == 0) {
#pragma unroll
            for (int r = 0; r < 8; ++r) pdot[wave][r] = acc[r];
        }
    }
#else
    // Host-pass parse placeholder (never executed; kernel body not emitted for host).
    if (lane == 0) {
        for (int r = 0; r < 8; ++r) {
            float d = 0.f;
            for (int c = 0; c < 128; ++c) d += tile[r][wave * 128 + c] * qw[wave * 128 + c];
            pdot[wave][r] = d;
        }
    }
#endif
    __syncthreads();

    // ---------------- Phase 3: logits -> softmax over l (tiny serial section) --------
    if (tid == 0) {
        float lg[8], mx = -3.4e38f;
#pragma unroll
        for (int l = 0; l < 8; ++l) {
            float d = 0.f;
#pragma unroll
            for (int w = 0; w < 8; ++w) d += pdot[w][l];
            const float inv = rsqrtf(sumsq[l] * (1.f / (float)DQ) + 1e-6f);
            lg[l] = d * inv;
            mx = fmaxf(mx, lg[l]);
        }
        float se = 0.f;
#pragma unroll
        for (int l = 0; l < 8; ++l) { lg[l] = expf(lg[l] - mx); se += lg[l]; }
        const float r = 1.f / se;
#pragma unroll
        for (int l = 0; l < 8; ++l) attn[l] = lg[l] * r;
    }
    __syncthreads();

    // ---------------- Phase 4: h[d] = sum_l attn[l] * V[l,d] -------------------------
    {
        float4 h = make_float4(0.f, 0.f, 0.f, 0.f);
        const int d0 = tid * 4;
#pragma unroll
        for (int l = 0; l < 8; ++l) {
            const float  a = attn[l];
            const float4 v = *(const float4*)&tile[l][d0];
            h.x += a * v.x; h.y += a * v.y; h.z += a * v.z; h.w += a * v.w;
        }
        ((float4*)out)[(size_t)gid * (DQ / 4) + tid] = h;
    }
}

extern "C" void kernel_launch(void* const* d_in, const int* in_sizes, int n_in,
                              void* d_out, int out_size, void* d_ws, size_t ws_size,
                              hipStream_t stream) {
    (void)in_sizes; (void)n_in; (void)out_size; (void)d_ws; (void)ws_size;
    const float* blocks = (const float*)d_in[0];
    const float* query  = (const float*)d_in[1];
    const float* normw  = (const float*)d_in[2];
    float*       out    = (float*)d_out;

    dim3 grid(BQ * TQ);   // one workgroup per (b, t)
    dim3 block(256);      // 8 wave32s
    hipLaunchKernelGGL(BlockAttentionResidual_22591527977642_kernel,
                       grid, block, 0, stream, blocks, query, normw, out);
}